// ADSR_7782480740867
// MI455X (gfx1250) — compile-verified
//
#include <hip/hip_runtime.h>
#include <math.h>

// ---------------------------------------------------------------------------
// ADSR envelope, segment-decomposed.
// gate transitions partition [0,T) into alternating on/off segments.
// All cross-segment coupling (2-pass refinement in the reference) reduces to
// closed-form functions of a handful of neighboring transition indices.
// ---------------------------------------------------------------------------

#define BLK   256
#define SCANB 1024               // threads in the single-block scan
#define MAXT  (1 << 23)          // supports up to 8M samples (harness uses 4M)
#define MAXNB (MAXT / BLK)       // 32768 blocks max

__device__ unsigned g_blk_cnt[MAXNB];
__device__ unsigned g_ex_cnt[MAXNB];
__device__ unsigned g_nseg;
__device__ int      g_seg_start[MAXT];
__device__ float    g_segP[MAXT];

__device__ __forceinline__ bool is_trans(const float* gate, int t) {
    if (t == 0) return true;
    return gate[t] != gate[t - 1];
}

// ---- K1: per-block transition counts (wave32 ballot + popc) ----------------
__global__ void k_count(const float* __restrict__ gate, int n) {
    int t = blockIdx.x * BLK + threadIdx.x;
    // streaming prefetch hint (lowers to global_prefetch_b8 on gfx1250)
    __builtin_prefetch(gate + t + 8 * BLK, 0, 1);
    bool tr = (t < n) ? is_trans(gate, t) : false;
    unsigned m = __builtin_amdgcn_ballot_w32(tr);
    __shared__ unsigned wcnt[BLK / 32];
    unsigned lane = threadIdx.x & 31u;
    unsigned w    = threadIdx.x >> 5;
    if (lane == 0) wcnt[w] = __popc(m);
    __syncthreads();
    if (threadIdx.x == 0) {
        unsigned s = 0;
        for (int i = 0; i < BLK / 32; ++i) s += wcnt[i];
        g_blk_cnt[blockIdx.x] = s;
    }
}

// ---- K2: parallel exclusive scan of block counts (one 1024-thread block) ---
__global__ void k_scan(int nb) {
    __shared__ unsigned partial[SCANB];
    int tid = threadIdx.x;
    int per = (nb + SCANB - 1) / SCANB;
    int s   = tid * per;
    int e   = s + per; if (e > nb) e = nb;

    unsigned sum = 0;
    for (int i = s; i < e; ++i) sum += g_blk_cnt[i];
    partial[tid] = sum;
    __syncthreads();

    // Hillis-Steele inclusive scan over 1024 partials (10 steps)
    unsigned val = sum;
    for (int off = 1; off < SCANB; off <<= 1) {
        unsigned t = (tid >= off) ? partial[tid - off] : 0u;
        __syncthreads();
        val += t;
        partial[tid] = val;
        __syncthreads();
    }
    unsigned acc = val - sum;                 // exclusive prefix of this chunk
    for (int i = s; i < e; ++i) { g_ex_cnt[i] = acc; acc += g_blk_cnt[i]; }
    if (tid == SCANB - 1) g_nseg = val;       // inclusive total
}

// ---- K3: scatter transition positions into seg_start[] ---------------------
__global__ void k_write(const float* __restrict__ gate, int n) {
    int t = blockIdx.x * BLK + threadIdx.x;
    bool tr = (t < n) ? is_trans(gate, t) : false;
    unsigned m    = __builtin_amdgcn_ballot_w32(tr);
    unsigned lane = threadIdx.x & 31u;
    unsigned w    = threadIdx.x >> 5;
    unsigned before = __popc(m & ((1u << lane) - 1u));
    __shared__ unsigned wcnt[BLK / 32];
    if (lane == 0) wcnt[w] = __popc(m);
    __syncthreads();
    unsigned wex = 0;
    for (unsigned i = 0; i < w; ++i) wex += wcnt[i];
    if (tr) g_seg_start[g_ex_cnt[blockIdx.x] + wex + before] = t;
}

// ---- K4: per-segment closed-form 2-pass parameters --------------------------
struct Params { float attack, decay, sustain, release; };

__device__ __forceinline__ int seg_end(unsigned k, unsigned nseg, int n) {
    return (k + 1 < nseg) ? g_seg_start[k + 1] : n;
}
__device__ __forceinline__ float seg_len(unsigned k, unsigned nseg, int n) {
    return (float)(seg_end(k, nseg, n) - g_seg_start[k]);
}
__device__ __forceinline__ float dval(float L, const Params& P) {
    // decay value at axis == L (note-off during decay)
    return P.sustain + (1.f - P.sustain) * expf(-(L - P.attack) / P.decay);
}
__device__ __forceinline__ float noteoff1(unsigned k, unsigned nseg, int n, const Params& P) {
    float L = seg_len(k, nseg, n);
    return (L > P.attack) ? dval(L, P) : L / P.attack;   // pass-1: rstart = 0
}
__device__ __forceinline__ float rstart1(unsigned k, unsigned nseg, int n, const Params& P) {
    if (k < 2) return 0.f;                               // no prior on-segment
    float gap = seg_len(k - 1, nseg, n);                 // preceding off-run
    return noteoff1(k - 2, nseg, n, P) * expf(-gap / P.release);
}
__device__ __forceinline__ float noteoff2(unsigned k, unsigned nseg, int n, const Params& P) {
    float L = seg_len(k, nseg, n);
    if (L > P.attack) return dval(L, P);
    float r1 = rstart1(k, nseg, n, P);
    return r1 + (1.f - r1) * (L / P.attack);
}
__device__ __forceinline__ float rstart2(unsigned k, unsigned nseg, int n, const Params& P) {
    if (k < 2) return 0.f;
    float gap = seg_len(k - 1, nseg, n);
    return noteoff2(k - 2, nseg, n, P) * expf(-gap / P.release);
}

__global__ void k_seg(const float* __restrict__ gate, int n,
                      const float* pa, const float* pd, const float* ps, const float* pr) {
    unsigned k = blockIdx.x * BLK + threadIdx.x;
    unsigned nseg = g_nseg;
    if (k >= nseg) return;
    Params P{*pa, *pd, *ps, *pr};
    bool on = gate[g_seg_start[k]] > 0.5f;
    float p;
    if (on)          p = rstart2(k, nseg, n, P);                 // attack start level
    else if (k >= 1) p = noteoff2(k - 1, nseg, n, P);            // release start level
    else             p = 0.f;                                    // leading silence
    g_segP[k] = p;
}

// ---- K5: per-sample emit, 4 elems/thread (b128 loads/stores) ----------------
__global__ void k_emit(const float* __restrict__ gate, float* __restrict__ out, int n,
                       const float* pa, const float* pd, const float* ps, const float* pr) {
    int base = (blockIdx.x * BLK + threadIdx.x) * 4;
    if (base >= n) return;
    __builtin_prefetch(gate + base + 32 * BLK, 0, 1);
    float attack = *pa, decay = *pd, sustain = *ps, release = *pr;

    unsigned nseg = g_nseg;
    unsigned lo = 0, hi = nseg - 1;
    while (lo < hi) {                       // largest k with seg_start[k] <= base
        unsigned mid = (lo + hi + 1) >> 1;
        if (g_seg_start[mid] <= base) lo = mid; else hi = mid - 1;
    }

    int  rem  = n - base;
    bool full = rem >= 4;
    int  cnt  = full ? 4 : rem;

    float gv[4];
    if (full) {
        float4 g4 = *(const float4*)(gate + base);
        gv[0] = g4.x; gv[1] = g4.y; gv[2] = g4.z; gv[3] = g4.w;
    } else {
        for (int i = 0; i < cnt; ++i) gv[i] = gate[base + i];
    }

    float res[4];
    for (int i = 0; i < cnt; ++i) {
        int t = base + i;
        while (lo + 1 < nseg && g_seg_start[lo + 1] <= t) ++lo;  // rare crossings
        int   s = g_seg_start[lo];
        float p = g_segP[lo];
        float j = (float)(t - s + 1);       // 1-indexed position in segment
        float v;
        if (gv[i] > 0.5f) {
            if (j <= attack) v = p + (1.f - p) * (j / attack);                        // attack
            else             v = sustain + (1.f - sustain) * expf(-(j - attack) / decay); // decay
        } else {
            v = p * expf(-j / release);                                               // release
        }
        res[i] = v;
    }

    if (full) *(float4*)(out + base) = make_float4(res[0], res[1], res[2], res[3]);
    else      for (int i = 0; i < cnt; ++i) out[base + i] = res[i];
}

// ---------------------------------------------------------------------------
extern "C" void kernel_launch(void* const* d_in, const int* in_sizes, int n_in,
                              void* d_out, int out_size, void* d_ws, size_t ws_size,
                              hipStream_t stream) {
    (void)n_in; (void)d_ws; (void)ws_size; (void)out_size;
    const float* gate = (const float*)d_in[0];
    const float* pa   = (const float*)d_in[1];
    const float* pd   = (const float*)d_in[2];
    const float* ps   = (const float*)d_in[3];
    const float* pr   = (const float*)d_in[4];
    float* out = (float*)d_out;

    int n = in_sizes[0];
    if (n <= 0) return;
    if (n > MAXT) n = MAXT;                 // static scratch bound (harness: 4M)
    int nb  = (n + BLK - 1) / BLK;
    int n4  = (n + 3) / 4;
    int nbe = (n4 + BLK - 1) / BLK;

    k_count<<<nb,  BLK,   0, stream>>>(gate, n);
    k_scan <<<1,   SCANB, 0, stream>>>(nb);
    k_write<<<nb,  BLK,   0, stream>>>(gate, n);
    k_seg  <<<nb,  BLK,   0, stream>>>(gate, n, pa, pd, ps, pr);
    k_emit <<<nbe, BLK,   0, stream>>>(gate, out, n, pa, pd, ps, pr);
}